// S4Layer_48172353192499
// MI455X (gfx1250) — compile-verified
//
#include <hip/hip_runtime.h>

typedef float v2f __attribute__((ext_vector_type(2)));
typedef float v8f __attribute__((ext_vector_type(8)));

#define TWO_PI 6.28318530717958647692f

__device__ __forceinline__ v8f wmma4(v2f a, v2f b, v8f c) {
  return __builtin_amdgcn_wmma_f32_16x16x4_f32(false, a, false, b, (short)0, c,
                                               false, false);
}

// DFT matrix fragments in WMMA A-layout: element (m,k) at
// lane = m + 16*h with h=(k>>1)&1, vgpr j=k&1, chunk c=k>>2. r=cos, i=sin, ni=-sin.
struct DF { v2f r[4]; v2f i[4]; v2f ni[4]; };

__device__ __forceinline__ void make_d16(DF& d, float sgn, int lane) {
  int m = lane & 15, h = (lane >> 4) & 1;
#pragma unroll
  for (int c = 0; c < 4; ++c) {
#pragma unroll
    for (int j = 0; j < 2; ++j) {
      int k = 4 * c + 2 * h + j;
      float ang = sgn * (TWO_PI / 16.0f) * (float)((m * k) & 15);
      float s, co; __sincosf(ang, &s, &co);
      d.r[c][j] = co; d.i[c][j] = s; d.ni[c][j] = -s;
    }
  }
}

// block-diag(DFT8, DFT8) for packed dual FFT-128 second stage
__device__ __forceinline__ void make_bd8(DF& d, float sgn, int lane) {
  int m = lane & 15, h = (lane >> 4) & 1;
#pragma unroll
  for (int c = 0; c < 4; ++c) {
#pragma unroll
    for (int j = 0; j < 2; ++j) {
      int k = 4 * c + 2 * h + j;
      bool on = ((m >> 3) == (k >> 3));
      float ang = sgn * (TWO_PI / 8.0f) * (float)(((m & 7) * (k & 7)) & 7);
      float s, co; __sincosf(ang, &s, &co);
      d.r[c][j] = on ? co : 0.f;
      d.i[c][j] = on ? s : 0.f;
      d.ni[c][j] = on ? -s : 0.f;
    }
  }
}

// ---------------------------------------------------------------------------
// Wave-level 256-point FFT (radix 16x16) via v_wmma_f32_16x16x4_f32.
// in/out are LDS element pointers with a stride (in float2 units).
// scr: per-wave 16*17 float2 LDS scratch (mid transpose).
// If EXT: multiply output q by exp(SGN*2pi*i * (j*q mod Nbig)/Nbig).
// ---------------------------------------------------------------------------
template <int SGN, bool EXT>
__device__ __forceinline__ void fft256_wave(const float2* __restrict__ in, int inS,
                                            float2* __restrict__ out, int outS,
                                            float2* __restrict__ scr, const DF& d,
                                            int lane, int jcol, int nbigMask,
                                            float wNbig) {
  const int lm = lane & 15;
  const int h = (lane >> 4) & 1;

  // B-layout fragments of V[m2][m1] = v[m1 + 16*m2]
  v2f vr[4], vi[4];
#pragma unroll
  for (int c = 0; c < 4; ++c) {
    int k0 = 4 * c + 2 * h;
    float2 e0 = in[(lm + 16 * k0) * inS];
    float2 e1 = in[(lm + 16 * (k0 + 1)) * inS];
    vr[c][0] = e0.x; vr[c][1] = e1.x;
    vi[c][0] = e0.y; vi[c][1] = e1.y;
  }

  // A = D16 x V  (complex, 16 wmma)
  v8f cr = {0.f,0.f,0.f,0.f,0.f,0.f,0.f,0.f};
  v8f ci = {0.f,0.f,0.f,0.f,0.f,0.f,0.f,0.f};
#pragma unroll
  for (int c = 0; c < 4; ++c) cr = wmma4(d.r[c], vr[c], cr);
#pragma unroll
  for (int c = 0; c < 4; ++c) cr = wmma4(d.ni[c], vi[c], cr);
#pragma unroll
  for (int c = 0; c < 4; ++c) ci = wmma4(d.r[c], vi[c], ci);
#pragma unroll
  for (int c = 0; c < 4; ++c) ci = wmma4(d.i[c], vr[c], ci);

  // mid twiddle W256^{m1*k2} then transpose through LDS scratch
#pragma unroll
  for (int v = 0; v < 8; ++v) {
    int k2 = v + 8 * h;
    int m1 = lm;
    float ang = (float)SGN * (TWO_PI / 256.0f) * (float)(m1 * k2);
    float s, co; __sincosf(ang, &s, &co);
    float ar = cr[v], ai = ci[v];
    scr[m1 * 17 + k2] = make_float2(ar * co - ai * s, ar * s + ai * co);
  }
  asm volatile("s_wait_dscnt 0x0" ::: "memory");

  v2f tr[4], ti[4];
#pragma unroll
  for (int c = 0; c < 4; ++c) {
    int k0 = 4 * c + 2 * h;
    float2 e0 = scr[k0 * 17 + lm];
    float2 e1 = scr[(k0 + 1) * 17 + lm];
    tr[c][0] = e0.x; tr[c][1] = e1.x;
    ti[c][0] = e0.y; ti[c][1] = e1.y;
  }

  // out = D16 x A'^T (16 wmma)
  v8f orr = {0.f,0.f,0.f,0.f,0.f,0.f,0.f,0.f};
  v8f oii = {0.f,0.f,0.f,0.f,0.f,0.f,0.f,0.f};
#pragma unroll
  for (int c = 0; c < 4; ++c) orr = wmma4(d.r[c], tr[c], orr);
#pragma unroll
  for (int c = 0; c < 4; ++c) orr = wmma4(d.ni[c], ti[c], orr);
#pragma unroll
  for (int c = 0; c < 4; ++c) oii = wmma4(d.r[c], ti[c], oii);
#pragma unroll
  for (int c = 0; c < 4; ++c) oii = wmma4(d.i[c], tr[c], oii);

#pragma unroll
  for (int v = 0; v < 8; ++v) {
    int q = lm + 16 * (v + 8 * h);
    float re = orr[v], im = oii[v];
    if (EXT) {
      int t = (jcol * q) & nbigMask;
      float s, co; __sincosf(wNbig * (float)t, &s, &co);
      float r2 = re * co - im * s;
      im = re * s + im * co; re = r2;
    }
    out[q * outS] = make_float2(re, im);
  }
}

// ---------------------------------------------------------------------------
// Wave-level PAIR of 128-point FFTs (radix 16x8), packed in one 16x16 tile.
// inRow0 points at first row (128 float2); second row at inRow0+rowS.
// ---------------------------------------------------------------------------
template <int SGN>
__device__ __forceinline__ void fft128pair_wave(const float2* __restrict__ inRow0,
                                                int rowS, float2* __restrict__ outRow0,
                                                int orowS, float2* __restrict__ scr,
                                                const DF& d16, const DF& bd8, int lane) {
  const int lm = lane & 15;
  const int h = (lane >> 4) & 1;
  const int fid = lm >> 3;
  const int m1 = lm & 7;

  v2f vr[4], vi[4];
#pragma unroll
  for (int c = 0; c < 4; ++c) {
    int k0 = 4 * c + 2 * h;
    float2 e0 = inRow0[fid * rowS + m1 + 8 * k0];
    float2 e1 = inRow0[fid * rowS + m1 + 8 * (k0 + 1)];
    vr[c][0] = e0.x; vr[c][1] = e1.x;
    vi[c][0] = e0.y; vi[c][1] = e1.y;
  }

  v8f cr = {0.f,0.f,0.f,0.f,0.f,0.f,0.f,0.f};
  v8f ci = {0.f,0.f,0.f,0.f,0.f,0.f,0.f,0.f};
#pragma unroll
  for (int c = 0; c < 4; ++c) cr = wmma4(d16.r[c], vr[c], cr);
#pragma unroll
  for (int c = 0; c < 4; ++c) cr = wmma4(d16.ni[c], vi[c], cr);
#pragma unroll
  for (int c = 0; c < 4; ++c) ci = wmma4(d16.r[c], vi[c], ci);
#pragma unroll
  for (int c = 0; c < 4; ++c) ci = wmma4(d16.i[c], vr[c], ci);

#pragma unroll
  for (int v = 0; v < 8; ++v) {
    int k2 = v + 8 * h;
    float ang = (float)SGN * (TWO_PI / 128.0f) * (float)(m1 * k2);
    float s, co; __sincosf(ang, &s, &co);
    float ar = cr[v], ai = ci[v];
    scr[lm * 17 + k2] = make_float2(ar * co - ai * s, ar * s + ai * co);
  }
  asm volatile("s_wait_dscnt 0x0" ::: "memory");

  v2f tr[4], ti[4];
#pragma unroll
  for (int c = 0; c < 4; ++c) {
    int k0 = 4 * c + 2 * h;
    float2 e0 = scr[k0 * 17 + lm];
    float2 e1 = scr[(k0 + 1) * 17 + lm];
    tr[c][0] = e0.x; tr[c][1] = e1.x;
    ti[c][0] = e0.y; ti[c][1] = e1.y;
  }

  v8f orr = {0.f,0.f,0.f,0.f,0.f,0.f,0.f,0.f};
  v8f oii = {0.f,0.f,0.f,0.f,0.f,0.f,0.f,0.f};
#pragma unroll
  for (int c = 0; c < 4; ++c) orr = wmma4(bd8.r[c], tr[c], orr);
#pragma unroll
  for (int c = 0; c < 4; ++c) orr = wmma4(bd8.ni[c], ti[c], orr);
#pragma unroll
  for (int c = 0; c < 4; ++c) oii = wmma4(bd8.r[c], ti[c], oii);
#pragma unroll
  for (int c = 0; c < 4; ++c) oii = wmma4(bd8.i[c], tr[c], oii);

#pragma unroll
  for (int v = 0; v < 8; ++v) {
    int r = v + 8 * h;
    int ofid = r >> 3, k1 = r & 7;
    int q = lm + 16 * k1;
    outRow0[ofid * orowS + q] = make_float2(orr[v], oii[v]);
  }
}

// ============================ small kernels ================================

__global__ void k_setup(const float* __restrict__ Ctp, const float* __restrict__ Bp,
                        const float* __restrict__ ls, const float2* __restrict__ p,
                        const float2* __restrict__ q, const float2* __restrict__ Vc,
                        float2* __restrict__ wprod, float* __restrict__ stepOut) {
  int n = threadIdx.x;  // 64
  float br = 0.f, bi = 0.f;
  for (int m = 0; m < 64; ++m) {
    float2 v = Vc[n * 64 + m];
    float b = Bp[m];
    br += v.x * b; bi += v.y * b;
  }
  float a0r = Ctp[2 * n], a0i = -Ctp[2 * n + 1];          // conj(Ct)
  float2 qq = q[n]; float a1r = qq.x, a1i = -qq.y;        // conj(q)
  float2 pp = p[n]; float b1r = pp.x, b1i = pp.y;
  wprod[n]       = make_float2(a0r * br - a0i * bi, a0r * bi + a0i * br);
  wprod[64 + n]  = make_float2(a0r * b1r - a0i * b1i, a0r * b1i + a0i * b1r);
  wprod[128 + n] = make_float2(a1r * br - a1i * bi, a1r * bi + a1i * br);
  wprod[192 + n] = make_float2(a1r * b1r - a1i * b1i, a1r * b1i + a1i * b1r);
  if (n == 0) *stepOut = __expf(ls[0]);
}

__global__ __launch_bounds__(256) void k_cauchy(const float2* __restrict__ lambda,
                                                const float2* __restrict__ wprod,
                                                const float* __restrict__ stepP,
                                                float2* __restrict__ A) {
  __shared__ float2 sL[64], s00[64], s01[64], s10[64], s11[64];
  int tid = threadIdx.x;
  if (tid < 64) {
    sL[tid] = lambda[tid];
    s00[tid] = wprod[tid]; s01[tid] = wprod[64 + tid];
    s10[tid] = wprod[128 + tid]; s11[tid] = wprod[192 + tid];
  }
  __syncthreads();
  int l = blockIdx.x * 256 + tid;
  float step = *stepP;
  float s, c;
  __sincosf(-TWO_PI * (float)l / 32768.0f, &s, &c);   // Omega
  float opr = 1.f + c, opi = s;
  float omr = 1.f - c, omi = -s;
  float inv = 1.f / (opr * opr + opi * opi);
  float cfr = 2.f * opr * inv, cfi = -2.f * opi * inv;  // 2/(1+Om)
  float ts = 2.f / step;
  float gr = ts * (omr * opr + omi * opi) * inv;
  float gi = ts * (omi * opr - omr * opi) * inv;        // g
  float k00r=0,k00i=0,k01r=0,k01i=0,k10r=0,k10i=0,k11r=0,k11i=0;
  for (int n = 0; n < 64; ++n) {
    float dr = gr - sL[n].x, di = gi - sL[n].y;
    float idn = 1.f / (dr * dr + di * di);
    float er = dr * idn, ei = -di * idn;                // 1/(g-lambda)
    float2 w;
    w = s00[n]; k00r += w.x*er - w.y*ei; k00i += w.x*ei + w.y*er;
    w = s01[n]; k01r += w.x*er - w.y*ei; k01i += w.x*ei + w.y*er;
    w = s10[n]; k10r += w.x*er - w.y*ei; k10i += w.x*ei + w.y*er;
    w = s11[n]; k11r += w.x*er - w.y*ei; k11i += w.x*ei + w.y*er;
  }
  float pr = 1.f + k11r, pi = k11i;
  float inv2 = 1.f / (pr * pr + pi * pi);
  float tr_ = k01r * k10r - k01i * k10i, ti_ = k01r * k10i + k01i * k10r;
  float ur = (tr_ * pr + ti_ * pi) * inv2, ui = (ti_ * pr - tr_ * pi) * inv2;
  float vr_ = k00r - ur, vi_ = k00i - ui;
  float atr = cfr * vr_ - cfi * vi_, ati = cfr * vi_ + cfi * vr_;
  const float sc = 1.0f / 32768.0f;                     // fold iFFT-32k scale
  A[l] = make_float2(atr * sc, ati * sc);
}

// ===================== 32768-point inverse FFT of atRoots ===================
// K[t1+256*t2] = sum_{k1<128} e^{+2pi i t1 k1/32768} e^{+2pi i t2 k1/128}
//               * sum_{k2<256} A[k1+128*k2] e^{+2pi i t1 k2/256}

__global__ __launch_bounds__(256) void k_ifft32k_p1(const float2* __restrict__ A,
                                                    float2* __restrict__ G1) {
  __shared__ float2 tin[256 * 9];
  __shared__ float2 tout[256 * 9];
  __shared__ float2 scr[8 * 272];
  int tid = threadIdx.x;
  int k1base = blockIdx.x * 8;
#pragma unroll
  for (int it = 0; it < 8; ++it) {
    int r = (tid >> 3) + 32 * it, col = tid & 7;        // r = k2
    tin[r * 9 + col] = A[(k1base + col) + 128 * r];
  }
  __syncthreads();
  int wave = tid >> 5, lane = tid & 31;
  DF d; make_d16(d, 1.f, lane);
  fft256_wave<1, true>(&tin[wave], 9, &tout[wave], 9, &scr[wave * 272], d, lane,
                       k1base + wave, 32767, TWO_PI / 32768.0f);
  __syncthreads();
#pragma unroll
  for (int it = 0; it < 8; ++it) {
    int r = (tid >> 3) + 32 * it, col = tid & 7;        // r = t1
    G1[r * 128 + (k1base + col)] = tout[r * 9 + col];
  }
}

__global__ __launch_bounds__(256) void k_ifft32k_p2(const float2* __restrict__ G1,
                                                    float* __restrict__ Kpad) {
  __shared__ float2 tin[16 * 129];
  __shared__ float2 tout[16 * 129];
  __shared__ float2 scr[8 * 272];
  int tid = threadIdx.x;
  int t1base = blockIdx.x * 16;
#pragma unroll
  for (int it = 0; it < 8; ++it) {
    int e = tid + 256 * it;
    int r = e >> 7, c = e & 127;
    tin[r * 129 + c] = G1[(t1base + r) * 128 + c];
  }
  __syncthreads();
  int wave = tid >> 5, lane = tid & 31;
  DF d16; make_d16(d16, 1.f, lane);
  DF bd8; make_bd8(bd8, 1.f, lane);
  fft128pair_wave<1>(&tin[wave * 2 * 129], 129, &tout[wave * 2 * 129], 129,
                     &scr[wave * 272], d16, bd8, lane);
  __syncthreads();
#pragma unroll
  for (int it = 0; it < 8; ++it) {
    int r = tid & 15;                 // t1 local (fast for coalescing)
    int t2 = (tid >> 4) + 16 * it;    // 0..127
    Kpad[(t1base + r) + 256 * t2] = tout[r * 129 + t2].x;
  }
}

// ======================= 65536-point forward FFT ===========================
// x = ua + i*ub, zero padded. Pass1: per column n1, FFT256 over n2, twiddle,
// store G[k2*256+n1]. Pass2: per row k2, FFT256 over n1 in place -> Zp[k2*256+k1].

__global__ __launch_bounds__(256) void k_fwd_p1(const float* __restrict__ xbase,
                                                float2* __restrict__ Gbase) {
  __shared__ float2 tin[256 * 9];
  __shared__ float2 tout[256 * 9];
  __shared__ float2 scr[8 * 272];
  const float* ua = xbase + (size_t)blockIdx.y * 65536;
  const float* ub = ua + 32768;
  float2* G = Gbase + (size_t)blockIdx.y * 65536;
  int tid = threadIdx.x;
  int n1base = blockIdx.x * 8;
#pragma unroll
  for (int it = 0; it < 8; ++it) {
    int n2 = (tid >> 3) + 32 * it, col = tid & 7;
    float re = 0.f, im = 0.f;
    if (n2 < 128) {
      int idx = (n1base + col) + 256 * n2;
      re = ua[idx]; im = ub[idx];
    }
    tin[n2 * 9 + col] = make_float2(re, im);
  }
  __syncthreads();
  int wave = tid >> 5, lane = tid & 31;
  DF d; make_d16(d, -1.f, lane);
  fft256_wave<-1, true>(&tin[wave], 9, &tout[wave], 9, &scr[wave * 272], d, lane,
                        n1base + wave, 65535, -TWO_PI / 65536.0f);
  __syncthreads();
#pragma unroll
  for (int it = 0; it < 8; ++it) {
    int k2 = (tid >> 3) + 32 * it, col = tid & 7;
    G[k2 * 256 + (n1base + col)] = tout[k2 * 9 + col];
  }
}

__global__ __launch_bounds__(256) void k_fwd_p2(float2* __restrict__ Gbase) {
  __shared__ float2 tin[16 * 257];
  __shared__ float2 scr[8 * 272];
  float2* G = Gbase + (size_t)blockIdx.y * 65536;
  int tid = threadIdx.x;
  int k2base = blockIdx.x * 16;
#pragma unroll
  for (int it = 0; it < 16; ++it) {
    int e = tid + 256 * it;
    int r = e >> 8, c = e & 255;
    tin[r * 257 + c] = G[(k2base + r) * 256 + c];
  }
  __syncthreads();
  int wave = tid >> 5, lane = tid & 31;
  DF d; make_d16(d, -1.f, lane);
#pragma unroll
  for (int t = 0; t < 2; ++t) {
    int rl = wave * 2 + t;
    fft256_wave<-1, false>(&tin[rl * 257], 1, &tin[rl * 257], 1, &scr[wave * 272],
                           d, lane, 0, 0, 0.f);
  }
  __syncthreads();
#pragma unroll
  for (int it = 0; it < 16; ++it) {
    int e = tid + 256 * it;
    int r = e >> 8, c = e & 255;
    G[(k2base + r) * 256 + c] = tin[r * 257 + c];
  }
}

// ===================== 65536-point inverse + fused Kd ======================
// Pass1: per row k2: Y = Z*Kd/65536, invFFT256 over k1, twiddle, H[m1*256+k2].
// Pass2: per row m1: invFFT256 over k2; y[m1+256*m2] = re/im + D*u.

__global__ __launch_bounds__(256) void k_inv_p1(const float2* __restrict__ Gbase,
                                                const float2* __restrict__ Kdp,
                                                float2* __restrict__ Hbase) {
  __shared__ float2 tin[8 * 257];
  __shared__ float2 tout[256 * 9];
  __shared__ float2 scr[8 * 272];
  const float2* Z = Gbase + (size_t)blockIdx.y * 65536;
  float2* H = Hbase + (size_t)blockIdx.y * 65536;
  int tid = threadIdx.x;
  int k2base = blockIdx.x * 8;
  const float sc = 1.0f / 65536.0f;
#pragma unroll
  for (int it = 0; it < 8; ++it) {
    int e = tid + 256 * it;
    int r = e >> 8, c = e & 255;
    float2 z = Z[(k2base + r) * 256 + c];
    float2 kd = Kdp[(k2base + r) * 256 + c];
    tin[r * 257 + c] = make_float2((z.x * kd.x - z.y * kd.y) * sc,
                                   (z.x * kd.y + z.y * kd.x) * sc);
  }
  __syncthreads();
  int wave = tid >> 5, lane = tid & 31;
  DF d; make_d16(d, 1.f, lane);
  fft256_wave<1, true>(&tin[wave * 257], 1, &tout[wave], 9, &scr[wave * 272], d,
                       lane, k2base + wave, 65535, TWO_PI / 65536.0f);
  __syncthreads();
#pragma unroll
  for (int it = 0; it < 8; ++it) {
    int m1 = (tid >> 3) + 32 * it, col = tid & 7;
    H[m1 * 256 + (k2base + col)] = tout[m1 * 9 + col];
  }
}

__global__ __launch_bounds__(256) void k_inv_p2(const float2* __restrict__ Hbase,
                                                const float* __restrict__ ubase,
                                                float* __restrict__ ybase,
                                                const float* __restrict__ Dptr) {
  __shared__ float2 tin[8 * 257];
  __shared__ float2 tout[256 * 9];
  __shared__ float2 scr[8 * 272];
  const float2* H = Hbase + (size_t)blockIdx.y * 65536;
  const float* ua = ubase + (size_t)blockIdx.y * 65536;
  const float* ub = ua + 32768;
  float* ya = ybase + (size_t)blockIdx.y * 65536;
  float* yb = ya + 32768;
  int tid = threadIdx.x;
  int m1base = blockIdx.x * 8;
#pragma unroll
  for (int it = 0; it < 8; ++it) {
    int e = tid + 256 * it;
    int r = e >> 8, c = e & 255;
    tin[r * 257 + c] = H[(m1base + r) * 256 + c];
  }
  __syncthreads();
  int wave = tid >> 5, lane = tid & 31;
  DF d; make_d16(d, 1.f, lane);
  fft256_wave<1, false>(&tin[wave * 257], 1, &tout[wave], 9, &scr[wave * 272], d,
                        lane, 0, 0, 0.f);
  __syncthreads();
  float Dv = Dptr[0];
#pragma unroll
  for (int it = 0; it < 4; ++it) {
    int m2 = (tid >> 3) + 32 * it;   // 0..127 only (t < 32768)
    int col = tid & 7;
    int tpos = (m1base + col) + 256 * m2;
    float2 v = tout[m2 * 9 + col];
    ya[tpos] = v.x + Dv * ua[tpos];
    yb[tpos] = v.y + Dv * ub[tpos];
  }
}

// ================================ host =====================================

extern "C" void kernel_launch(void* const* d_in, const int* in_sizes, int n_in,
                              void* d_out, int out_size, void* d_ws, size_t ws_size,
                              hipStream_t stream) {
  (void)in_sizes; (void)n_in; (void)out_size;
  const float* u   = (const float*)d_in[0];   // (256, 32768)
  const float* Ctp = (const float*)d_in[1];   // (1,64,2)
  const float* Bp  = (const float*)d_in[2];   // (64,1)
  const float* ls  = (const float*)d_in[3];   // (1,)
  const float* Dv  = (const float*)d_in[4];   // (1,)
  const float2* Lam = (const float2*)d_in[5]; // (64,) c64
  const float2* p   = (const float2*)d_in[6]; // (64,) c64
  const float2* q   = (const float2*)d_in[7]; // (64,) c64
  const float2* Vc  = (const float2*)d_in[8]; // (64,64) c64
  float* y = (float*)d_out;

  char* ws = (char*)d_ws;
  float2* wprod = (float2*)(ws);                       // 256 f2
  float*  stepP = (float*)(ws + 2048);
  float2* A     = (float2*)(ws + 4096);                // 32768 f2
  float2* G1    = (float2*)(ws + 4096 + 262144);       // 32768 f2
  float*  Kpad  = (float*)(ws + 4096 + 524288);        // 65536 f (upper half 0)
  float2* KdG   = (float2*)(ws + 4096 + 786432);       // 65536 f2
  const size_t FIXED = 4096 + 786432 + 524288;         // 1,314,816 B
  const size_t perPair = 2ull * 65536ull * sizeof(float2);  // G + H
  int CH = 1;
  if (ws_size > FIXED + perPair) {
    size_t c = (ws_size - FIXED) / perPair;
    CH = (c > 128) ? 128 : (int)c;
    if (CH < 1) CH = 1;
  }
  float2* BufG = (float2*)(ws + FIXED);
  float2* BufH = BufG + (size_t)CH * 65536;

  (void)hipMemsetAsync(Kpad, 0, 65536 * sizeof(float), stream);
  k_setup<<<1, 64, 0, stream>>>(Ctp, Bp, ls, p, q, Vc, wprod, stepP);
  k_cauchy<<<128, 256, 0, stream>>>(Lam, wprod, stepP, A);
  k_ifft32k_p1<<<16, 256, 0, stream>>>(A, G1);
  k_ifft32k_p2<<<16, 256, 0, stream>>>(G1, Kpad);
  // spectrum of K (real, zero-padded): fft(K + i*0) using the pair pipeline
  k_fwd_p1<<<dim3(32, 1), 256, 0, stream>>>(Kpad, KdG);
  k_fwd_p2<<<dim3(16, 1), 256, 0, stream>>>(KdG);
  // batch pairs: z = ifft(fft(u2b + i*u2b1) * Kd); re->row 2b, im->row 2b+1
  for (int c0 = 0; c0 < 128; c0 += CH) {
    int n = 128 - c0; if (n > CH) n = CH;
    const float* uc = u + (size_t)c0 * 65536;
    float* yc = y + (size_t)c0 * 65536;
    k_fwd_p1<<<dim3(32, n), 256, 0, stream>>>(uc, BufG);
    k_fwd_p2<<<dim3(16, n), 256, 0, stream>>>(BufG);
    k_inv_p1<<<dim3(32, n), 256, 0, stream>>>(BufG, KdG, BufH);
    k_inv_p2<<<dim3(32, n), 256, 0, stream>>>(BufH, uc, yc, Dv);
  }
}